// LinearCapsPro_82660940579043
// MI455X (gfx1250) — compile-verified
//
#include <hip/hip_runtime.h>
#include <hip/hip_bf16.h>
#include <math.h>

typedef __attribute__((ext_vector_type(2))) float v2f;
typedef __attribute__((ext_vector_type(8))) float v8f;

#define CP_EPS  1e-4f
#define CP_B    4096
#define CP_F    2048
#define CP_C    100
#define CP_D    16

// ---------------------------------------------------------------------------
// Kernel 1 (setup, one wave32 per capsule):
//   1) Gram G = W_c W_c^T via chained V_WMMA_F32_16X16X4_F32 (A-frag==B-frag)
//   2) Cholesky  G + eps*I = L L^T  (16x16, in LDS)
//   3) V_c = L^{-1} W_c  by forward substitution, 64 columns per lane
// Math: u^T inv(G+eps I) u = ||L^{-1} u||^2 = ||V_c x||^2, so the main kernel
// needs no sigma at all -- its epilogue is a row-wise sum of squares.
// ---------------------------------------------------------------------------
__global__ __launch_bounds__(32) void capspro_setup_kernel(
    const float* __restrict__ W,   // [C, D, F]
    float* __restrict__ V)         // [C, D, F] out
{
    const int c    = blockIdx.x;
    const int l    = threadIdx.x;
    const int half = l >> 4;
    const int lr   = l & 15;

    // --- Gram via WMMA: A-frag == B-frag (same W_c rows) ---
    const float* wrow = W + ((size_t)c * CP_D + lr) * CP_F + half * 2;
    v8f acc = {};
#pragma unroll 8
    for (int k = 0; k < CP_F; k += 4) {
        v2f a = *(const v2f*)(wrow + k);
        acc = __builtin_amdgcn_wmma_f32_16x16x4_f32(
            false, a, false, a, (short)0, acc, false, false);
    }

    __shared__ float M[16][17];    // col 16 later holds 1/L[d][d]
#pragma unroll
    for (int j = 0; j < 8; ++j) {
        const int m = j + half * 8;
        M[m][lr] = acc[j] + (m == lr ? CP_EPS : 0.0f);
    }
    __syncthreads();

    // --- Cholesky (right-looking), lower triangle in place ---
    for (int k = 0; k < 16; ++k) {
        const float dk = sqrtf(M[k][k]);
        __syncthreads();
        if (l == k)                M[k][k] = dk;
        else if (l < 16 && l > k)  M[l][k] /= dk;
        __syncthreads();
        if (l > k && l < 16) {
            const float lik = M[l][k];
            for (int j = k + 1; j <= l; ++j) M[l][j] -= lik * M[j][k];
        }
        __syncthreads();
    }
    if (l < 16) M[l][16] = 1.0f / M[l][l];   // reciprocal diagonal in the pad
    __syncthreads();

    // --- forward substitution: solve L v = w for every feature column ---
    const float* wc = W + (size_t)c * CP_D * CP_F;
    float*       vc = V + (size_t)c * CP_D * CP_F;
    for (int f = l; f < CP_F; f += 32) {        // 64 columns per lane, coalesced
        float v[16];
#pragma unroll
        for (int d = 0; d < 16; ++d) {
            float s = wc[(size_t)d * CP_F + f];
#pragma unroll
            for (int j = 0; j < d; ++j) s -= M[d][j] * v[j];
            v[d] = s * M[d][16];
        }
#pragma unroll
        for (int d = 0; d < 16; ++d) vc[(size_t)d * CP_F + f] = v[d];
    }
}

// ---------------------------------------------------------------------------
// Kernel 2: register-blocked GEMM  v = X @ V_c^T, out = sqrt(rowsum(v^2)).
// One wave32 handles 4 batch tiles (64 rows) x 4 capsules:
//   16 independent f32 WMMA chains, 8 b64 loads feed 16 WMMAs per K-step.
//   L2 traffic 3.36GB*(1/4+1/4) = 1.68 GB.
// Epilogue: 8 squares + 4 shfl_xor reductions per tile -- no LDS, no sigma.
// Grid = (4096/64) x (100/4) = 64 x 25.
// ---------------------------------------------------------------------------
__global__ __launch_bounds__(32) void capspro_main_kernel(
    const float* __restrict__ x,      // [B, F]
    const float* __restrict__ V,      // [C, D, F]
    float* __restrict__ out)          // [B, C]
{
    const int rb   = blockIdx.x * 64;  // 4 row tiles of 16
    const int cb   = blockIdx.y * 4;   // 4 capsules
    const int l    = threadIdx.x;
    const int half = l >> 4;
    const int lr   = l & 15;

    const float* xp0 = x + ((size_t)(rb +  0 + lr)) * CP_F + half * 2;
    const float* xp1 = x + ((size_t)(rb + 16 + lr)) * CP_F + half * 2;
    const float* xp2 = x + ((size_t)(rb + 32 + lr)) * CP_F + half * 2;
    const float* xp3 = x + ((size_t)(rb + 48 + lr)) * CP_F + half * 2;
    const float* vp0 = V + ((size_t)(cb + 0) * CP_D + lr) * CP_F + half * 2;
    const float* vp1 = V + ((size_t)(cb + 1) * CP_D + lr) * CP_F + half * 2;
    const float* vp2 = V + ((size_t)(cb + 2) * CP_D + lr) * CP_F + half * 2;
    const float* vp3 = V + ((size_t)(cb + 3) * CP_D + lr) * CP_F + half * 2;

    v8f acc[4][4] = {};

#pragma unroll 2
    for (int k = 0; k < CP_F; k += 4) {
        v2f a[4], b[4];
        a[0] = *(const v2f*)(xp0 + k);
        a[1] = *(const v2f*)(xp1 + k);
        a[2] = *(const v2f*)(xp2 + k);
        a[3] = *(const v2f*)(xp3 + k);
        b[0] = *(const v2f*)(vp0 + k);
        b[1] = *(const v2f*)(vp1 + k);
        b[2] = *(const v2f*)(vp2 + k);
        b[3] = *(const v2f*)(vp3 + k);
#pragma unroll
        for (int i = 0; i < 4; ++i)
#pragma unroll
            for (int j = 0; j < 4; ++j)
                acc[i][j] = __builtin_amdgcn_wmma_f32_16x16x4_f32(
                    false, a[i], false, b[j], (short)0, acc[i][j], false, false);
    }

    // ---- epilogue: out[m, c] = sqrt( sum_n v[m][n]^2 ) ----
    // C-layout: VGPR q holds row m = q + 8*half, columns n = lane&15 striped
    // across the 16 lanes of each half -> xor-shuffle reduce within the half.
#pragma unroll
    for (int i = 0; i < 4; ++i) {
#pragma unroll
        for (int j = 0; j < 4; ++j) {
#pragma unroll
            for (int q = 0; q < 8; ++q) {
                float p = acc[i][j][q] * acc[i][j][q];
                p += __shfl_xor(p, 1);
                p += __shfl_xor(p, 2);
                p += __shfl_xor(p, 4);
                p += __shfl_xor(p, 8);
                if (lr == q) {
                    const int row = rb + i * 16 + q + 8 * half;
                    out[(size_t)row * CP_C + (cb + j)] = sqrtf(p);
                }
            }
        }
    }
}

extern "C" void kernel_launch(void* const* d_in, const int* in_sizes, int n_in,
                              void* d_out, int out_size, void* d_ws, size_t ws_size,
                              hipStream_t stream) {
    (void)in_sizes; (void)n_in; (void)out_size; (void)ws_size;
    const float* x = (const float*)d_in[0];   // [4096, 2048]
    const float* W = (const float*)d_in[1];   // [100, 16, 2048]
    float* out     = (float*)d_out;           // [4096, 100]
    float* V       = (float*)d_ws;            // [100, 16, 2048] f32 scratch (12.8 MB)

    capspro_setup_kernel<<<dim3(CP_C), dim3(32), 0, stream>>>(W, V);
    capspro_main_kernel<<<dim3(CP_B / 64, CP_C / 4), dim3(32), 0, stream>>>(x, V, out);
}